// EntanglementFormationModel_31619549233404
// MI455X (gfx1250) — compile-verified
//
#include <hip/hip_runtime.h>
#include <cstddef>

// ---------------------------------------------------------------------------
// EntanglementFormationModel for MI455X (gfx1250, wave32, WMMA).
// Pipeline: gram = theta^H theta  ->  Newton-Schulz inverse-sqrt (GEMM-only)
//        -> theta1 = theta @ gram^-1/2 -> prob -> psiAB = (theta1*sqrtEVL)@EVC^H
//        -> per-term 32x32 Hermitian Jacobi eigenvalues -> entropy -> dot.
// All heavy math runs through V_WMMA_F32_16X16X4_F32 (FP32 WMMA).
// ---------------------------------------------------------------------------

typedef float v2f __attribute__((ext_vector_type(2)));
typedef float v8f __attribute__((ext_vector_type(8)));

#define NTERM 4096
#define DDIM  1024
#define EPSV  1e-12f

// ---------------------------------------------------------------------------
// Complex GEMM: C = alpha*(op(A) op(B)) + gamma*I  (optionally alpha *= *dscale)
// op via OPA/OPB (0=N, 1=T), conjugation via CONJA/CONJB (sign composition).
// Block tile 128x64, 8 waves, 32x32 per wave, K step 16 (4 x k4 WMMA).
// B tile is stored K-transposed in LDS so both A and B fragments are
// contiguous (k, k+1) pairs -> single ds_load_2addr into even-aligned pairs,
// no repack movs in front of the WMMA group.
// ---------------------------------------------------------------------------
template<int OPA, int OPB, int CONJA, int CONJB>
__global__ void __launch_bounds__(256)
cgemm_wmma(const float* __restrict__ Ar, const float* __restrict__ Ai,
           const float* __restrict__ Br, const float* __restrict__ Bi,
           float* __restrict__ Cr, float* __restrict__ Ci,
           int M, int N, int K, int lda, int ldb, int ldc,
           float alpha, float gamma, const float* __restrict__ dscale)
{
    constexpr int BM = 128, BN = 64, BK = 16;
    __shared__ float Asr[BM][BK + 1], Asi[BM][BK + 1];   // op(A) tile, M-major
    __shared__ float Bsr[BN][BK + 1], Bsi[BN][BK + 1];   // op(B) tile, N-major (K-transposed)

    const int t    = threadIdx.x;
    const int bm   = blockIdx.y * BM, bn = blockIdx.x * BN;
    const int wid  = t >> 5, lane = t & 31;
    const int lm   = lane & 15, hi = lane >> 4;
    const int wm   = (wid & 3) * 32;   // 4 waves along M
    const int wn   = (wid >> 2) * 32;  // 2 waves along N

    __builtin_amdgcn_s_wait_tensorcnt(0);

    v8f acc[2][2][4];
    for (int mt = 0; mt < 2; ++mt)
        for (int nt = 0; nt < 2; ++nt)
            for (int p = 0; p < 4; ++p)
                for (int e = 0; e < 8; ++e) acc[mt][nt][p][e] = 0.f;

    for (int kk = 0; kk < K; kk += BK) {
        // ---- stage op(A) tile: BM x BK (M-major) ----
        if (OPA == 0) {
            for (int idx = t; idx < BM * BK; idx += 256) {
                int c = idx & (BK - 1), r = idx / BK;     // consecutive t -> consecutive k
                Asr[r][c] = Ar[(size_t)(bm + r) * lda + kk + c];
                Asi[r][c] = Ai[(size_t)(bm + r) * lda + kk + c];
            }
        } else {
            for (int idx = t; idx < BM * BK; idx += 256) {
                int r = idx & (BM - 1), c = idx / BM;     // consecutive t -> consecutive m
                Asr[r][c] = Ar[(size_t)(kk + c) * lda + bm + r];
                Asi[r][c] = Ai[(size_t)(kk + c) * lda + bm + r];
            }
        }
        // ---- stage op(B) tile: BK x BN, stored as [BN][BK] ----
        if (OPB == 0) {
            for (int idx = t; idx < BK * BN; idx += 256) {
                int c = idx & (BN - 1), r = idx / BN;     // consecutive t -> consecutive n
                Bsr[c][r] = Br[(size_t)(kk + r) * ldb + bn + c];
                Bsi[c][r] = Bi[(size_t)(kk + r) * ldb + bn + c];
            }
        } else {
            for (int idx = t; idx < BK * BN; idx += 256) {
                int r = idx & (BK - 1), c = idx / BK;     // consecutive t -> consecutive k
                Bsr[c][r] = Br[(size_t)(bn + c) * ldb + kk + r];
                Bsi[c][r] = Bi[(size_t)(bn + c) * ldb + kk + r];
            }
        }
        // stream the next K-tile into cache while we compute this one
        if (kk + BK < K) {
            __builtin_prefetch(&Ar[OPA == 0 ? (size_t)(bm + (t >> 1)) * lda + kk + BK
                                            : (size_t)(kk + BK + (t & 15)) * lda + bm], 0, 1);
            __builtin_prefetch(&Br[OPB == 0 ? (size_t)(kk + BK + (t & 15)) * ldb + bn
                                            : (size_t)(bn + (t & 63)) * ldb + kk + BK], 0, 1);
        }
        __syncthreads();

#pragma unroll
        for (int k0 = 0; k0 < BK; k0 += 4) {
            v2f arf[2], aif[2], brf[2], bif[2];
#pragma unroll
            for (int mt = 0; mt < 2; ++mt) {
                int m = wm + mt * 16 + lm;      // A 16x4: lanes 0-15 K={0,1}, 16-31 K={2,3}
                arf[mt][0] = Asr[m][k0 + 2 * hi];
                arf[mt][1] = Asr[m][k0 + 2 * hi + 1];
                aif[mt][0] = Asi[m][k0 + 2 * hi];
                aif[mt][1] = Asi[m][k0 + 2 * hi + 1];
            }
#pragma unroll
            for (int nt = 0; nt < 2; ++nt) {
                int n = wn + nt * 16 + lm;      // B 4x16: N = lane&15, K split across halves
                brf[nt][0] = Bsr[n][k0 + 2 * hi];
                brf[nt][1] = Bsr[n][k0 + 2 * hi + 1];
                bif[nt][0] = Bsi[n][k0 + 2 * hi];
                bif[nt][1] = Bsi[n][k0 + 2 * hi + 1];
            }
#pragma unroll
            for (int mt = 0; mt < 2; ++mt)
#pragma unroll
                for (int nt = 0; nt < 2; ++nt) {
                    acc[mt][nt][0] = __builtin_amdgcn_wmma_f32_16x16x4_f32(
                        false, arf[mt], false, brf[nt], (short)0, acc[mt][nt][0], false, false);
                    acc[mt][nt][1] = __builtin_amdgcn_wmma_f32_16x16x4_f32(
                        false, aif[mt], false, bif[nt], (short)0, acc[mt][nt][1], false, false);
                    acc[mt][nt][2] = __builtin_amdgcn_wmma_f32_16x16x4_f32(
                        false, arf[mt], false, bif[nt], (short)0, acc[mt][nt][2], false, false);
                    acc[mt][nt][3] = __builtin_amdgcn_wmma_f32_16x16x4_f32(
                        false, aif[mt], false, brf[nt], (short)0, acc[mt][nt][3], false, false);
                }
        }
        __syncthreads();
    }

    float scale = alpha;
    if (dscale) scale *= dscale[0];
    const float sA = CONJA ? -1.f : 1.f, sB = CONJB ? -1.f : 1.f;
    // (ar + i*sA*ai)(br + i*sB*bi) = (rr - sA*sB*ii) + i*(sA*ir + sB*ri)
#pragma unroll
    for (int mt = 0; mt < 2; ++mt)
#pragma unroll
        for (int nt = 0; nt < 2; ++nt)
#pragma unroll
            for (int v = 0; v < 8; ++v) {
                int row = bm + wm + mt * 16 + v + 8 * hi;   // C/D: VGPR v -> M = v / v+8
                int col = bn + wn + nt * 16 + lm;
                float cr = scale * (acc[mt][nt][0][v] - sA * sB * acc[mt][nt][1][v]);
                float ci = scale * (sB * acc[mt][nt][2][v] + sA * acc[mt][nt][3][v]);
                if (gamma != 0.f && row == col) cr += gamma;
                Cr[(size_t)row * ldc + col] = cr;
                Ci[(size_t)row * ldc + col] = ci;
            }
}

// ---------------------------------------------------------------------------
// trace of gram -> [0]=1/tr, [1]=1/sqrt(tr)  (Newton-Schulz normalization)
// ---------------------------------------------------------------------------
__global__ void k_trace(const float* __restrict__ Gr, float* __restrict__ scal)
{
    __shared__ float red[256];
    float s = 0.f;
    for (int i = threadIdx.x; i < DDIM; i += 256) s += Gr[(size_t)i * DDIM + i];
    red[threadIdx.x] = s; __syncthreads();
    for (int o = 128; o > 0; o >>= 1) {
        if (threadIdx.x < o) red[threadIdx.x] += red[threadIdx.x + o];
        __syncthreads();
    }
    if (threadIdx.x == 0) {
        float tr = fmaxf(red[0], 1e-30f);
        scal[0] = 1.f / tr;
        scal[1] = rsqrtf(tr);
    }
}

__global__ void k_init_YZ(const float* __restrict__ Gr, const float* __restrict__ Gi,
                          float* __restrict__ Yr, float* __restrict__ Yi,
                          float* __restrict__ Zr, float* __restrict__ Zi,
                          const float* __restrict__ scal)
{
    const float is_ = scal[0];
    const long total = (long)DDIM * DDIM;
    for (long i = (long)blockIdx.x * 256 + threadIdx.x; i < total; i += (long)gridDim.x * 256) {
        Yr[i] = Gr[i] * is_;
        Yi[i] = Gi[i] * is_;
        int r = (int)(i >> 10), c = (int)(i & (DDIM - 1));
        Zr[i] = (r == c) ? 1.f : 0.f;
        Zi[i] = 0.f;
    }
}

// prob[n] = sum_j |theta1[n][j]|^2 * EVL[j]
__global__ void k_prob(const float* __restrict__ Tr, const float* __restrict__ Ti,
                       const float* __restrict__ EVL, float* __restrict__ prob)
{
    __shared__ float red[256];
    const int n = blockIdx.x;
    float s = 0.f;
    for (int j = threadIdx.x; j < DDIM; j += 256) {
        float a = Tr[(size_t)n * DDIM + j], b = Ti[(size_t)n * DDIM + j];
        s += (a * a + b * b) * EVL[j];
    }
    red[threadIdx.x] = s; __syncthreads();
    for (int o = 128; o > 0; o >>= 1) {
        if (threadIdx.x < o) red[threadIdx.x] += red[threadIdx.x + o];
        __syncthreads();
    }
    if (threadIdx.x == 0) prob[n] = red[0];
}

// in-place column scale by sqrt(EVL[j])
__global__ void k_scale_cols(float* __restrict__ Tr, float* __restrict__ Ti,
                             const float* __restrict__ EVL, long total)
{
    for (long i = (long)blockIdx.x * 256 + threadIdx.x; i < total; i += (long)gridDim.x * 256) {
        int j = (int)(i & (DDIM - 1));
        float s = sqrtf(fmaxf(EVL[j], 0.f));
        Tr[i] *= s;
        Ti[i] *= s;
    }
}

// ---------------------------------------------------------------------------
// Per-term: psi (32x32 complex, scaled by 1/sqrt(prob)), rdm = psi psi^H,
// eigenvalues via parallel cyclic complex Jacobi (tournament pairing,
// 16 independent rotations / round, 31 rounds / sweep), entropy = -sum l log l.
// One 256-thread workgroup per term.
// ---------------------------------------------------------------------------
__global__ void __launch_bounds__(256)
k_rdm_ent(const float* __restrict__ Pr, const float* __restrict__ Pi,
          const float* __restrict__ prob, float* __restrict__ ent)
{
    __shared__ float pr[32][33], pim[32][33];
    __shared__ float Hr[32][33], Hi[32][33];
    __shared__ float cA[16], srA[16], siA[16];
    __shared__ int   pA[16], qA[16], perm[32];
    __shared__ float red32[32];

    const int n = blockIdx.x, t = threadIdx.x;
    const float rn = rsqrtf(fmaxf(prob[n], 1e-30f));

    for (int e = t; e < 1024; e += 256) {
        int i = e >> 5, j = e & 31;
        pr[i][j]  = Pr[(size_t)n * 1024 + e] * rn;
        pim[i][j] = Pi[(size_t)n * 1024 + e] * rn;
    }
    if (t < 32) perm[t] = t;
    __syncthreads();

    // rdm[i][k] = sum_j psi[i][j] * conj(psi[k][j])
    for (int e = t; e < 1024; e += 256) {
        int i = e >> 5, k = e & 31;
        float hr = 0.f, hi2 = 0.f;
        for (int j = 0; j < 32; ++j) {
            float a = pr[i][j], b = pim[i][j], c = pr[k][j], d = pim[k][j];
            hr  += a * c + b * d;
            hi2 += b * c - a * d;
        }
        Hr[i][k] = hr; Hi[i][k] = hi2;
    }
    __syncthreads();

    for (int sweep = 0; sweep < 8; ++sweep) {
        for (int round = 0; round < 31; ++round) {
            if (t < 16) {
                int p = perm[t], q = perm[31 - t];
                if (p > q) { int tmp = p; p = q; q = tmp; }
                float app = Hr[p][p], aqq = Hr[q][q];
                float ar = Hr[p][q], ai2 = Hi[p][q];
                float mag = sqrtf(ar * ar + ai2 * ai2);
                float c = 1.f, s = 0.f, phr = 1.f, phi = 0.f;
                if (mag > 1e-24f) {
                    phr = ar / mag; phi = ai2 / mag;
                    float tau = (aqq - app) / (2.f * mag);
                    float tt = (tau >= 0.f ? 1.f : -1.f) / (fabsf(tau) + sqrtf(1.f + tau * tau));
                    c = rsqrtf(1.f + tt * tt);
                    s = tt * c;
                }
                cA[t] = c; srA[t] = s * phr; siA[t] = s * phi; pA[t] = p; qA[t] = q;
            }
            __syncthreads();
            // rows: A <- J^H A  (each work item owns rows p,q at one column)
            for (int w = t; w < 512; w += 256) {
                int pair = w >> 5, j = w & 31;
                int p = pA[pair], q = qA[pair];
                float c = cA[pair], sr = srA[pair], si = siA[pair];
                float apr = Hr[p][j], api = Hi[p][j];
                float aqr = Hr[q][j], aqi = Hi[q][j];
                Hr[p][j] = c * apr - (sr * aqr - si * aqi);
                Hi[p][j] = c * api - (sr * aqi + si * aqr);
                Hr[q][j] = (sr * apr + si * api) + c * aqr;
                Hi[q][j] = (sr * api - si * apr) + c * aqi;
            }
            __syncthreads();
            // cols: A <- A J
            for (int w = t; w < 512; w += 256) {
                int pair = w >> 5, i = w & 31;
                int p = pA[pair], q = qA[pair];
                float c = cA[pair], sr = srA[pair], si = siA[pair];
                float ipr = Hr[i][p], ipi = Hi[i][p];
                float iqr = Hr[i][q], iqi = Hi[i][q];
                Hr[i][p] = c * ipr - (sr * iqr + si * iqi);
                Hi[i][p] = c * ipi - (sr * iqi - si * iqr);
                Hr[i][q] = (sr * ipr - si * ipi) + c * iqr;
                Hi[i][q] = (sr * ipi + si * ipr) + c * iqi;
            }
            __syncthreads();
            if (t == 0) {  // rotate tournament (slot 0 fixed)
                int last = perm[31];
                for (int i = 31; i > 1; --i) perm[i] = perm[i - 1];
                perm[1] = last;
            }
            __syncthreads();
        }
    }

    if (t < 32) {
        float lam = Hr[t][t];
        red32[t] = -lam * __logf(fmaxf(lam, EPSV));
    }
    __syncthreads();
    if (t == 0) {
        float e = 0.f;
        for (int i = 0; i < 32; ++i) e += red32[i];
        ent[n] = e;
    }
}

__global__ void k_final(const float* __restrict__ prob, const float* __restrict__ ent,
                        float* __restrict__ out)
{
    __shared__ float red[256];
    float s = 0.f;
    for (int i = threadIdx.x; i < NTERM; i += 256) s += prob[i] * ent[i];
    red[threadIdx.x] = s; __syncthreads();
    for (int o = 128; o > 0; o >>= 1) {
        if (threadIdx.x < o) red[threadIdx.x] += red[threadIdx.x + o];
        __syncthreads();
    }
    if (threadIdx.x == 0) out[0] = red[0];
}

// ---------------------------------------------------------------------------
extern "C" void kernel_launch(void* const* d_in, const int* in_sizes, int n_in,
                              void* d_out, int out_size, void* d_ws, size_t ws_size,
                              hipStream_t stream)
{
    (void)in_sizes; (void)n_in; (void)out_size; (void)ws_size;
    const float* theta = (const float*)d_in[0];
    const float* thr = theta;                              // theta[0]: re (N x D)
    const float* thi = theta + (size_t)NTERM * DDIM;       // theta[1]: im
    const float* EVL  = (const float*)d_in[1];
    const float* EVCr = (const float*)d_in[2];
    const float* EVCi = (const float*)d_in[3];
    float* out = (float*)d_out;

    char* ws = (char*)d_ws;
    size_t off = 0;
    auto alloc = [&](size_t nf) { float* p = (float*)(ws + off); off += nf * sizeof(float); return p; };
    const size_t DD = (size_t)DDIM * DDIM, ND = (size_t)NTERM * DDIM;

    float* Gr = alloc(DD); float* Gi = alloc(DD);          // gram; reused as T after init
    float* Yr = alloc(DD); float* Yi = alloc(DD);
    float* Zr = alloc(DD); float* Zi = alloc(DD);
    float* Ur = alloc(DD); float* Ui = alloc(DD);
    float* Vr = alloc(DD); float* Vi = alloc(DD);
    float* T1r = alloc(ND); float* T1i = alloc(ND);        // theta1 (then scaled in-place)
    float* Pr  = alloc(ND); float* Pi  = alloc(ND);        // psiAB (unnormalized)
    float* prob = alloc(NTERM); float* ent = alloc(NTERM);
    float* scal = alloc(8);                                // [0]=1/tr, [1]=1/sqrt(tr)

    dim3 blk(256);
    dim3 gridD(DDIM / 64, DDIM / 128);    // 1024x1024 outputs
    dim3 gridN(DDIM / 64, NTERM / 128);   // 4096x1024 outputs

    // 1) gram = theta^H theta   (op(A)=conj-T over K=N)
    cgemm_wmma<1, 0, 1, 0><<<gridD, blk, 0, stream>>>(thr, thi, thr, thi, Gr, Gi,
        DDIM, DDIM, NTERM, DDIM, DDIM, DDIM, 1.f, 0.f, nullptr);

    // 2) trace normalization, Y0 = G/tr, Z0 = I
    k_trace<<<1, blk, 0, stream>>>(Gr, scal);
    k_init_YZ<<<4096, blk, 0, stream>>>(Gr, Gi, Yr, Yi, Zr, Zi, scal);

    // 3) Newton-Schulz: T = 1.5I - 0.5 Z Y ; Y <- Y T ; Z <- T Z
    float *yr = Yr, *yi = Yi, *zr = Zr, *zi = Zi;
    float *ur = Ur, *ui = Ui, *vr = Vr, *vi = Vi;
    float *tr = Gr, *ti = Gi;   // gram buffers free after init -> reuse for T
    for (int it = 0; it < 14; ++it) {
        cgemm_wmma<0, 0, 0, 0><<<gridD, blk, 0, stream>>>(zr, zi, yr, yi, tr, ti,
            DDIM, DDIM, DDIM, DDIM, DDIM, DDIM, -0.5f, 1.5f, nullptr);
        cgemm_wmma<0, 0, 0, 0><<<gridD, blk, 0, stream>>>(yr, yi, tr, ti, ur, ui,
            DDIM, DDIM, DDIM, DDIM, DDIM, DDIM, 1.f, 0.f, nullptr);
        cgemm_wmma<0, 0, 0, 0><<<gridD, blk, 0, stream>>>(tr, ti, zr, zi, vr, vi,
            DDIM, DDIM, DDIM, DDIM, DDIM, DDIM, 1.f, 0.f, nullptr);
        float* tp;
        tp = yr; yr = ur; ur = tp;  tp = yi; yi = ui; ui = tp;
        tp = zr; zr = vr; vr = tp;  tp = zi; zi = vi; vi = tp;
    }

    // 4) theta1 = theta @ (Z / sqrt(tr))   (device scalar folded into epilogue)
    cgemm_wmma<0, 0, 0, 0><<<gridN, blk, 0, stream>>>(thr, thi, zr, zi, T1r, T1i,
        NTERM, DDIM, DDIM, DDIM, DDIM, DDIM, 1.f, 0.f, scal + 1);

    // 5) prob = |theta1|^2 @ EVL
    k_prob<<<NTERM, blk, 0, stream>>>(T1r, T1i, EVL, prob);

    // 6) theta1 *= sqrt(EVL) columnwise (in place)
    k_scale_cols<<<8192, blk, 0, stream>>>(T1r, T1i, EVL, (long)ND);

    // 7) psiAB = theta1s @ conj(EVC)^T
    cgemm_wmma<0, 1, 0, 1><<<gridN, blk, 0, stream>>>(T1r, T1i, EVCr, EVCi, Pr, Pi,
        NTERM, DDIM, DDIM, DDIM, DDIM, DDIM, 1.f, 0.f, nullptr);

    // 8) per-term rdm + Jacobi eigenvalues + entropy
    k_rdm_ent<<<NTERM, blk, 0, stream>>>(Pr, Pi, prob, ent);

    // 9) result = dot(prob, ent)
    k_final<<<1, blk, 0, stream>>>(prob, ent, out);
}